// NeuSSampler_87385404604911
// MI455X (gfx1250) — compile-verified
//
#include <hip/hip_runtime.h>
#include <hip/hip_bf16.h>
#include <math.h>

// ---------------------------------------------------------------------------
// NeuS hierarchical importance sampler for MI455X (gfx1250, wave32).
//
// Kernel 1 (sampler): one thread per ray. Maintains sorted (sbin, sdf) pairs,
//   does 4 refinement steps: NeuS alphas -> weights -> CDF -> inverse-CDF
//   sampling of 17 bins -> stable backward in-place merge (reproduces
//   jnp.argsort(stable) + take_along_axis semantics exactly since both input
//   lists are sorted). Writes e_final (R x 128 f32, 64 MiB) into d_ws.
//
// Kernel 2 (expand): one wave per ray. Computes pts = origin + dir * e via
//   V_WMMA_F32_16X16X4_F32 rank-1 updates (A col0 = e-chunk, B row0 = dir,
//   C = origin broadcast), stages the 384-float ray tile in LDS, then drains
//   it with 3x GLOBAL_STORE_ASYNC_FROM_LDS_B128 per lane (512 B contiguous
//   per instruction across the wave). Output = 192 MiB => ~8.6 us floor at
//   23.3 TB/s; coalesced async stores keep us on that roofline.
// ---------------------------------------------------------------------------

#define EPS_W 1e-5f

__global__ void __launch_bounds__(128)
neus_sampler_kernel(const float* __restrict__ origins,
                    const float* __restrict__ dirs,
                    const float* __restrict__ nearv,
                    const float* __restrict__ farv,
                    float* __restrict__ e_out,   // R x 128
                    int R) {
  int r = blockIdx.x * blockDim.x + threadIdx.x;
  if (r >= R) return;

  const float ox = origins[3 * r + 0], oy = origins[3 * r + 1], oz = origins[3 * r + 2];
  const float dx = dirs[3 * r + 0],    dy = dirs[3 * r + 1],    dz = dirs[3 * r + 2];
  const float nr = nearv[r];
  const float fr = farv[r];
  const float range = fr - nr;

  float sb[129];    // sorted bin boundaries (spacing domain)
  float sd[128];    // sdf at bin starts, aligned with sb[0..n-2]
  float w[128];     // weights
  float cdf[129];   // sampling CDF
  float nb[17];     // newly sampled bins
  float nsd[16];    // sdf at new bin starts

  // ---- init: sbins = linspace(0,1,65); sdf at the 64 starts ----
  for (int i = 0; i <= 64; ++i) sb[i] = (float)i * (1.0f / 64.0f);
  for (int i = 0; i < 64; ++i) {
    float e  = nr + sb[i] * range;
    float px = ox + dx * e, py = oy + dy * e, pz = oz + dz * e;
    sd[i] = sqrtf(px * px + py * py + pz * pz) - 1.0f;
  }

  int n = 65;  // current number of bin boundaries
  for (int it = 0; it < 4; ++it) {
    const float inv_s = 64.0f * (float)(1 << it);

    // ---- render_alphas + transmittance + weights (n-1 weights, last = 0) ----
    const int na = n - 2;  // number of alphas
    float trans = 1.0f, wsum = 0.0f, prev_cos = 0.0f;
    for (int i = 0; i < na; ++i) {
      float delta = (sb[i + 1] - sb[i]) * range;
      float s0 = sd[i], s1 = sd[i + 1];
      float mid  = 0.5f * (s0 + s1);
      float cosv = (s1 - s0) / (delta + 1e-5f);
      float cc   = fminf(prev_cos, cosv);   // prev_cos is the *raw* previous cos
      prev_cos   = cosv;
      cc = fminf(fmaxf(cc, -1000.0f), 0.0f);
      float pe = mid - cc * delta * 0.5f;
      float ne = mid + cc * delta * 0.5f;
      float pc = 1.0f / (1.0f + expf(-pe * inv_s));
      float nc = 1.0f / (1.0f + expf(-ne * inv_s));
      float alpha = (pc - nc + 1e-5f) / (pc + 1e-5f);
      float wt = alpha * trans;
      w[i] = wt;
      wsum += wt;
      trans *= (1.0f - alpha + 1e-7f);
    }
    const int nw = n - 1;
    w[na] = 0.0f;  // appended zero weight

    // ---- pdf_sample: pad, normalize, cdf = min(1, cumsum), prepend 0 ----
    float padding = fmaxf(EPS_W - wsum, 0.0f);
    float addw = padding / (float)nw;
    float dsum = wsum + padding;
    cdf[0] = 0.0f;
    float run = 0.0f;
    for (int i = 0; i < nw; ++i) {
      run += (w[i] + addw) / dsum;
      cdf[i + 1] = fminf(run, 1.0f);
    }

    // ---- inverse-CDF sample 17 bins; u sorted -> linear searchsorted ----
    int idx = 0;
    for (int k = 0; k < 17; ++k) {
      float u = (float)k * (1.0f / 17.0f) + (1.0f / 34.0f);
      while (idx < n && cdf[idx] <= u) ++idx;  // side='right'
      int below = min(max(idx - 1, 0), n - 1);
      int above = min(idx, n - 1);
      float c0 = cdf[below], c1 = cdf[above];
      float b0 = sb[below],  b1 = sb[above];
      float dnm = c1 - c0;
      if (dnm < 1e-5f) dnm = 1.0f;
      float t = fminf(fmaxf((u - c0) / dnm, 0.0f), 1.0f);
      nb[k] = b0 + t * (b1 - b0);
    }

    // ---- sdf at the 16 new starts (consumed next iteration) ----
    if (it < 3) {
      for (int k = 0; k < 16; ++k) {
        float e  = nr + nb[k] * range;
        float px = ox + dx * e, py = oy + dy * e, pz = oz + dz * e;
        nsd[k] = sqrtf(px * px + py * py + pz * pz) - 1.0f;
      }
    }

    // ---- stable backward in-place merge of (start, sdf) pairs ----
    // A = sb[0..n-2] (with sd), B = nb[0..15] (with nsd). Ties: A first
    // (matches stable argsort of concat([A, B])).
    float endv = fmaxf(sb[n - 1], nb[16]);
    int i = n - 2, j = 15, kk = n + 14;
    while (j >= 0) {
      if (i >= 0 && sb[i] > nb[j]) {
        sb[kk] = sb[i];
        if (it < 3) sd[kk] = sd[i];
        --i;
      } else {
        sb[kk] = nb[j];
        if (it < 3) sd[kk] = nsd[j];
        --j;
      }
      --kk;
    }
    n += 16;
    sb[n - 1] = endv;
  }

  // ---- write euclidean positions of the final 128 starts ----
  float* ep = e_out + (size_t)r * 128;
  for (int jj = 0; jj < 128; ++jj) ep[jj] = nr + sb[jj] * range;
}

// ---------------------------------------------------------------------------
// Expansion via V_WMMA_F32_16X16X4_F32 rank-1 update, one wave per ray,
// LDS-staged and drained with async LDS->global b128 stores.
// ---------------------------------------------------------------------------
typedef float v2f __attribute__((ext_vector_type(2)));
typedef float v8f __attribute__((ext_vector_type(8)));

#define WAVES_PER_BLOCK 8

__global__ void __launch_bounds__(32 * WAVES_PER_BLOCK)
neus_expand_wmma_kernel(const float* __restrict__ origins,
                        const float* __restrict__ dirs,
                        const float* __restrict__ efin,  // R x 128
                        float* __restrict__ out,         // R x 128 x 3
                        int R) {
  // Wave-private LDS staging: 384 floats (one ray's 128x3 output) per wave.
  __shared__ float smem[WAVES_PER_BLOCK * 384];

  const int lane   = threadIdx.x & 31;
  const int waveIb = threadIdx.x >> 5;  // wave index within block
  const int wave   = (int)((blockIdx.x * blockDim.x + threadIdx.x) >> 5);
  if (wave >= R) return;  // wave-uniform: EXEC stays all-ones for WMMA
  const int r = wave;

  const float dxv = dirs[3 * r + 0], dyv = dirs[3 * r + 1], dzv = dirs[3 * r + 2];
  const float oxv = origins[3 * r + 0], oyv = origins[3 * r + 1], ozv = origins[3 * r + 2];

  const int col = lane & 15;  // N index for B/C/D row-striped layouts

  // B (4x16): VGPR0 lanes 0-15 = row 0 (the direction), lanes 16-31 = row 1 (0);
  //           VGPR1 = rows 2,3 (0).
  v2f b;
  b.x = (lane < 16) ? ((col == 0) ? dxv : (col == 1) ? dyv : (col == 2) ? dzv : 0.0f)
                    : 0.0f;
  b.y = 0.0f;

  // C (16x16): column c holds origin[c] for c<3 (broadcast down rows).
  float oc = (col == 0) ? oxv : (col == 1) ? oyv : (col == 2) ? ozv : 0.0f;
  v8f c;
#pragma unroll
  for (int v = 0; v < 8; ++v) c[v] = oc;

  const float* ep = efin + (size_t)r * 128;
  float* wlds = smem + waveIb * 384;

#pragma unroll
  for (int t = 0; t < 8; ++t) {
    // A (16x4, f32): VGPR0 = K0 (lanes 0-15, M=lane) / K2 (lanes 16-31, zero);
    //                VGPR1 = K1/K3 (zero). Column 0 = e-chunk.
    v2f a;
    a.x = (lane < 16) ? ep[t * 16 + lane] : 0.0f;
    a.y = 0.0f;

    // D = A x B + C : D[j, c] = e[16t+j]*dir[c] + origin[c]
    v8f d = __builtin_amdgcn_wmma_f32_16x16x4_f32(
        /*neg_a=*/false, a, /*neg_b=*/false, b,
        /*c_mod=*/(short)0, c, /*reuse_a=*/false, /*reuse_b=*/false);

    // D layout: lane n holds column n (n = lane&15); lanes 0-15 carry rows
    // 0..7 in d[0..7], lanes 16-31 carry rows 8..15. Stage columns 0..2.
    if (col < 3) {
      int rowbase = (lane < 16) ? 0 : 8;
#pragma unroll
      for (int v = 0; v < 8; ++v) {
        int row = rowbase + v;
        wlds[(t * 16 + row) * 3 + col] = d[v];
      }
    }
  }

  // Drain LDS -> global with async b128 stores: 96 float4 chunks per ray,
  // chunk (j*32 + lane) per lane => 512 B contiguous per instruction.
  // Async reads of LDS are not HW-ordered with the ds_store writes above:
  // wait for DScnt first. Region is wave-private, so no barrier needed.
  asm volatile("s_wait_dscnt 0x0" ::: "memory");

  const unsigned lbase = (unsigned)(uintptr_t)wlds;  // LDS byte offset of wave region
  const unsigned long long gbase =
      (unsigned long long)(uintptr_t)(out + (size_t)r * 384);

#pragma unroll
  for (int j = 0; j < 3; ++j) {
    unsigned off = (unsigned)((j * 32 + lane) * 16);
    unsigned long long gaddr = gbase + off;
    unsigned laddr = lbase + off;
    asm volatile("global_store_async_from_lds_b128 %0, %1, off"
                 :
                 : "v"(gaddr), "v"(laddr)
                 : "memory");
  }
  asm volatile("s_wait_asynccnt 0x0" ::: "memory");
}

extern "C" void kernel_launch(void* const* d_in, const int* in_sizes, int n_in,
                              void* d_out, int out_size, void* d_ws, size_t ws_size,
                              hipStream_t stream) {
  const float* origins = (const float*)d_in[0];
  const float* dirs    = (const float*)d_in[1];
  const float* nearv   = (const float*)d_in[2];
  const float* farv    = (const float*)d_in[3];
  float* out  = (float*)d_out;
  float* efin = (float*)d_ws;  // needs R*128*4 bytes = 64 MiB for R=131072

  const int R = in_sizes[0] / 3;

  // Kernel 1: per-ray sampler (one thread per ray).
  {
    int threads = 128;
    int blocks = (R + threads - 1) / threads;
    neus_sampler_kernel<<<blocks, threads, 0, stream>>>(origins, dirs, nearv, farv,
                                                        efin, R);
  }

  // Kernel 2: WMMA expansion (one wave32 per ray), async LDS->global drain.
  {
    int threads = 32 * WAVES_PER_BLOCK;
    int wpb = WAVES_PER_BLOCK;
    int blocks = (R + wpb - 1) / wpb;
    neus_expand_wmma_kernel<<<blocks, threads, 0, stream>>>(origins, dirs, efin,
                                                            out, R);
  }
}